// QuantumBranch_49675591745604
// MI455X (gfx1250) — compile-verified
//
#include <hip/hip_runtime.h>
#include <hip/hip_bf16.h>
#include <math.h>

typedef __attribute__((ext_vector_type(2))) float v2f;
typedef __attribute__((ext_vector_type(8))) float v8f;

// ---------------------------------------------------------------------------
// Setup kernel: build the fixed 16x16 circuit unitary U from `weights`,
// fold in the product-state phases (-i)^popcount(k), and write Re/Im of V^T
// into d_ws pre-swizzled into the WMMA f32 B-operand layout (4 K-chunks of
// 16x16x4, 2 VGPRs each, 32 lanes).
// ---------------------------------------------------------------------------
__global__ void qb_setup(const float* __restrict__ weights, float* __restrict__ ws) {
    if (threadIdx.x != 0 || blockIdx.x != 0) return;
    float ur[16][16], ui[16][16];
    for (int i = 0; i < 16; ++i)
        for (int j = 0; j < 16; ++j) { ur[i][j] = (i == j) ? 1.f : 0.f; ui[i][j] = 0.f; }

    const int n = 4;
    for (int l = 0; l < 2; ++l) {
        // Rot gates: RZ(omega) @ RY(theta) @ RZ(phi) per wire
        for (int w = 0; w < n; ++w) {
            float phi = weights[(l * 4 + w) * 3 + 0];
            float th  = weights[(l * 4 + w) * 3 + 1];
            float om  = weights[(l * 4 + w) * 3 + 2];
            float c = cosf(0.5f * th), s = sinf(0.5f * th);
            float apo = 0.5f * (phi + om), amo = 0.5f * (phi - om);
            float ca = cosf(apo), sa = sinf(apo);
            float cm = cosf(amo), sm = sinf(amo);
            // g00 = ep*c, g01 = -em*s, g10 = conj(em)*s, g11 = conj(ep)*c
            float g00r =  ca * c, g00i = -sa * c;
            float g01r = -cm * s, g01i = -sm * s;
            float g10r =  cm * s, g10i = -sm * s;
            float g11r =  ca * c, g11i =  sa * c;
            int bit = 8 >> w;  // wire 0 is MSB of 4-bit index
            for (int i0 = 0; i0 < 16; ++i0) {
                if (i0 & bit) continue;
                int i1 = i0 | bit;
                for (int cc = 0; cc < 16; ++cc) {
                    float ar = ur[i0][cc], ai = ui[i0][cc];
                    float br = ur[i1][cc], bi = ui[i1][cc];
                    ur[i0][cc] = g00r * ar - g00i * ai + g01r * br - g01i * bi;
                    ui[i0][cc] = g00r * ai + g00i * ar + g01r * bi + g01i * br;
                    ur[i1][cc] = g10r * ar - g10i * ai + g11r * br - g11i * bi;
                    ui[i1][cc] = g10r * ai + g10i * ar + g11r * bi + g11i * br;
                }
            }
        }
        // CNOT ring: control w, target (w+r)%4 ; row permutation
        int r = l % (n - 1) + 1;
        for (int w = 0; w < n; ++w) {
            int t = (w + r) % n;
            int bc = 8 >> w, bt = 8 >> t;
            for (int i = 0; i < 16; ++i) {
                if ((i & bc) && !(i & bt)) {
                    int j = i | bt;
                    for (int cc = 0; cc < 16; ++cc) {
                        float tr = ur[i][cc]; ur[i][cc] = ur[j][cc]; ur[j][cc] = tr;
                        float ti = ui[i][cc]; ui[i][cc] = ui[j][cc]; ui[j][cc] = ti;
                    }
                }
            }
        }
    }

    // V[j][k] = U[j][k] * (-i)^popcount(k); pack B = V^T in WMMA B-operand order.
    for (int t = 0; t < 4; ++t)
        for (int v = 0; v < 2; ++v)
            for (int l = 0; l < 32; ++l) {
                int N = l & 15;
                int K = 4 * t + v + ((l & 16) ? 2 : 0);
                int pc = __popc(K) & 3;
                float pr = (pc == 0) ? 1.f : ((pc == 2) ? -1.f : 0.f);
                float pi = (pc == 1) ? -1.f : ((pc == 3) ? 1.f : 0.f);
                float vr = ur[N][K] * pr - ui[N][K] * pi;
                float vi = ur[N][K] * pi + ui[N][K] * pr;
                ws[t * 64 + v * 32 + l]       = vr;   // Re(V^T) operand
                ws[256 + t * 64 + v * 32 + l] = vi;   // Im(V^T) operand
            }
}

// ---------------------------------------------------------------------------
// Main kernel: one wave32 handles a tile of 16 samples.
// ---------------------------------------------------------------------------
__global__ __launch_bounds__(256) void qb_main(
    const float* __restrict__ x, const float* __restrict__ W,
    const float* __restrict__ bvec, const float* __restrict__ gamma,
    const float* __restrict__ beta, const float* __restrict__ ws,
    float* __restrict__ out, int ntiles) {

    const int lane = threadIdx.x & 31;
    const int wave = (int)((blockIdx.x * blockDim.x + threadIdx.x) >> 5);
    if (wave >= ntiles) return;          // wave-uniform: EXEC stays all-ones
    const int half = lane >> 4;          // 0 or 1
    const int ln   = lane & 15;
    const int hb   = half << 4;

    // B operands (shared matrix), lane-swizzled by setup kernel
    v2f bRe[4], bIm[4];
#pragma unroll
    for (int t = 0; t < 4; ++t) {
        bRe[t].x = ws[t * 64 + lane];
        bRe[t].y = ws[t * 64 + 32 + lane];
        bIm[t].x = ws[256 + t * 64 + lane];
        bIm[t].y = ws[256 + t * 64 + 32 + lane];
    }

    // Per-lane slice of projection / layernorm params: outputs o = 4*ln .. 4*ln+3
    float4 Wrow[4];
#pragma unroll
    for (int j = 0; j < 4; ++j) Wrow[j] = ((const float4*)W)[ln * 4 + j];
    float4 bq = ((const float4*)bvec)[ln];
    float4 gq = ((const float4*)gamma)[ln];
    float4 be = ((const float4*)beta)[ln];
    float bqa[4] = {bq.x, bq.y, bq.z, bq.w};
    float gqa[4] = {gq.x, gq.y, gq.z, gq.w};
    float bea[4] = {be.x, be.y, be.z, be.w};

    // Product-state magnitudes m_k for this lane's 8 K slots
    const int s = wave * 16 + ln;
    float4 xv = ((const float4*)x)[s];
    const float PI_2 = 1.57079632679f;
    float h0 = tanhf(xv.x) * PI_2, h1 = tanhf(xv.y) * PI_2;
    float h2 = tanhf(xv.z) * PI_2, h3 = tanhf(xv.w) * PI_2;
    float c0 = __cosf(h0), s0 = __sinf(h0);
    float c1 = __cosf(h1), s1 = __sinf(h1);
    float c2 = __cosf(h2), s2 = __sinf(h2);
    float c3 = __cosf(h3), s3 = __sinf(h3);
    float ph[4] = {c0 * c1, c0 * s1, s0 * c1, s0 * s1};  // high bits (wires 0,1) = k>>2
    float w2 = half ? s2 : c2;                            // low bits: lane half picks k&2
    float pla = w2 * c3, plb = w2 * s3;                   // k&1 = 0 / 1
    v2f A[4];
#pragma unroll
    for (int t = 0; t < 4; ++t) { A[t].x = ph[t] * pla; A[t].y = ph[t] * plb; }

    // D = m (16x16) x V^T : two real 16x16x16 products, K=16 via 4x K=4 WMMA
    v8f dRe = {0.f, 0.f, 0.f, 0.f, 0.f, 0.f, 0.f, 0.f};
    v8f dIm = {0.f, 0.f, 0.f, 0.f, 0.f, 0.f, 0.f, 0.f};
#pragma unroll
    for (int t = 0; t < 4; ++t) {
        dRe = __builtin_amdgcn_wmma_f32_16x16x4_f32(false, A[t], false, bRe[t],
                                                    (short)0, dRe, false, false);
        dIm = __builtin_amdgcn_wmma_f32_16x16x4_f32(false, A[t], false, bIm[t],
                                                    (short)0, dIm, false, false);
    }

    // Each accumulator row r: lanes 0-15 = sample r, lanes 16-31 = sample r+8,
    // lane index within half = basis state j.
#pragma unroll
    for (int r = 0; r < 8; ++r) {
        float re = dRe[r], im = dIm[r];
        float p = re * re + im * im;  // probability of basis state (ln) for my sample
        // Walsh-Hadamard transform across the 16-lane half: H[i]=sum_j (-1)^{i&j} p_j
#pragma unroll
        for (int t = 0; t < 4; ++t) {
            int m = 1 << t;
            float o = __shfl_xor(p, m, 32);
            p = (lane & m) ? (o - p) : (p + o);
        }
        // z_w = H[bit of wire w]; wire0 is MSB -> index 8
        float z0 = __shfl(p, hb | 8, 32);
        float z1 = __shfl(p, hb | 4, 32);
        float z2 = __shfl(p, hb | 2, 32);
        float z3 = __shfl(p, hb | 1, 32);
        // softmax over 4
        float mx = fmaxf(fmaxf(z0, z1), fmaxf(z2, z3));
        float e0 = __expf(z0 - mx), e1 = __expf(z1 - mx);
        float e2 = __expf(z2 - mx), e3 = __expf(z3 - mx);
        float inv = 1.f / (e0 + e1 + e2 + e3);
        float q0 = e0 * inv, q1 = e1 * inv, q2 = e2 * inv, q3 = e3 * inv;
        // projection: this lane produces outputs 4*ln..4*ln+3
        float hv[4], lsum = 0.f, lsq = 0.f;
#pragma unroll
        for (int j = 0; j < 4; ++j) {
            float v = q0 * Wrow[j].x + q1 * Wrow[j].y + q2 * Wrow[j].z +
                      q3 * Wrow[j].w + bqa[j];
            hv[j] = v; lsum += v; lsq += v * v;
        }
        // mean/var over 64 = 16 lanes x 4 values
#pragma unroll
        for (int t = 0; t < 4; ++t) {
            int m = 1 << t;
            lsum += __shfl_xor(lsum, m, 32);
            lsq  += __shfl_xor(lsq, m, 32);
        }
        float mean = lsum * (1.f / 64.f);
        float var  = lsq * (1.f / 64.f) - mean * mean;
        float rstd = rsqrtf(var + 1e-5f);
        int srow = wave * 16 + r + half * 8;
        float4 o4;
        o4.x = (hv[0] - mean) * rstd * gqa[0] + bea[0];
        o4.y = (hv[1] - mean) * rstd * gqa[1] + bea[1];
        o4.z = (hv[2] - mean) * rstd * gqa[2] + bea[2];
        o4.w = (hv[3] - mean) * rstd * gqa[3] + bea[3];
        ((float4*)out)[srow * 16 + ln] = o4;
    }
}

// ---------------------------------------------------------------------------
extern "C" void kernel_launch(void* const* d_in, const int* in_sizes, int n_in,
                              void* d_out, int out_size, void* d_ws, size_t ws_size,
                              hipStream_t stream) {
    (void)n_in; (void)out_size; (void)ws_size;
    const float* x       = (const float*)d_in[0];
    const float* weights = (const float*)d_in[1];
    const float* W       = (const float*)d_in[2];
    const float* b       = (const float*)d_in[3];
    const float* gamma   = (const float*)d_in[4];
    const float* beta    = (const float*)d_in[5];
    float* out = (float*)d_out;
    float* ws  = (float*)d_ws;

    int nsamples = in_sizes[0] / 4;
    int ntiles   = nsamples / 16;

    qb_setup<<<1, 1, 0, stream>>>(weights, ws);

    const int waves_per_block = 8;   // 256 threads = 8 wave32
    int blocks = (ntiles + waves_per_block - 1) / waves_per_block;
    qb_main<<<blocks, 256, 0, stream>>>(x, W, b, gamma, beta, ws, out, ntiles);
}